// LocalOperator_33432025432230
// MI455X (gfx1250) — compile-verified
//
#include <hip/hip_runtime.h>
#include <hip/hip_bf16.h>

// Problem constants (match reference setup_inputs: B=64, N=512, H=2048)
#define Bcfg 64
#define Ncfg 512
#define Hcfg 2048

typedef float v2f __attribute__((ext_vector_type(2)));
typedef float v8f __attribute__((ext_vector_type(8)));

// ---------------------------------------------------------------------------
// Kernel 1: cs[k*H+h] = (cosh(2*W), sinh(2*W))  -- 1M elements, one-time table
// ---------------------------------------------------------------------------
__global__ __launch_bounds__(256)
void rbm_coshsinh_kernel(const float* __restrict__ W, float2* __restrict__ cs) {
    int i = blockIdx.x * 256 + threadIdx.x;
    if (i < Ncfg * Hcfg) {
        float w2 = 2.0f * W[i];
        cs[i] = make_float2(coshf(w2), sinhf(w2));
    }
}

// ---------------------------------------------------------------------------
// Kernel 2: theta = x@W + b via V_WMMA_F32_16X16X4_F32, epilogue T = tanh(theta)
// One wave per 16x16 output tile. M=64 -> 4 tiles, N=H=2048 -> 128 tiles.
// Grid: 128 blocks x 128 threads (4 waves/block) = 512 tiles.
// Lane layouts per CDNA5 ISA 7.12.2:
//   A 16x4 f32:  lanes hold M=lane&15; VGPR0=K(kk+2*half), VGPR1=K(kk+2*half+1)
//   B 4x16 f32:  lanes hold N=lane&15; lanes16-31 = second half of K
//   C/D 16x16:   VGPR v -> row M = v + 8*half, col N = lane&15
// ---------------------------------------------------------------------------
__global__ __launch_bounds__(128)
void rbm_theta_tanh_kernel(const float* __restrict__ x, const float* __restrict__ W,
                           const float* __restrict__ bias, float* __restrict__ T) {
    const int lane = threadIdx.x & 31;
    const int wave = threadIdx.x >> 5;
    const int tile = blockIdx.x * 4 + wave;   // 0..511
    const int mt   = tile >> 7;               // 0..3   (M tile)
    const int nt   = tile & 127;              // 0..127 (N tile)
    const int half = lane >> 4;
    const int ml   = lane & 15;
    const int m    = mt * 16 + ml;            // A row for this lane
    const int n    = nt * 16 + ml;            // B/C column for this lane

    v8f acc = {};
    const float* __restrict__ xr = x + (size_t)m * Ncfg;
    for (int kk = 0; kk < Ncfg; kk += 4) {
        const int k0 = kk + 2 * half;
        v2f A, Bv;
        A.x  = xr[k0];
        A.y  = xr[k0 + 1];
        Bv.x = W[(size_t)k0 * Hcfg + n];
        Bv.y = W[(size_t)(k0 + 1) * Hcfg + n];
        acc = __builtin_amdgcn_wmma_f32_16x16x4_f32(
            /*neg_a=*/false, A, /*neg_b=*/false, Bv,
            /*c_mod=*/(short)0, acc, /*reuse_a=*/false, /*reuse_b=*/false);
    }
    const float bb = bias[n];
#pragma unroll
    for (int v = 0; v < 8; ++v) {
        const int row = mt * 16 + v + 8 * half;
        T[(size_t)row * Hcfg + n] = tanhf(acc[v] + bb);
    }
}

// ---------------------------------------------------------------------------
// Kernel 3: out[b] = sum_k Oxy[k] * exp(-2*x[b,k]*a[k])
//                      * prod_h ( cosh(2W_kh) - x[b,k]*tanh(theta[b,h])*sinh(2W_kh) )
// Grid (8, 64): block = 256 threads (8 waves); block owns (b, 64-wide k chunk).
// tanh(theta[b,:]) staged in LDS (8 KB); wave-per-k; lane product -> shfl reduce.
// ---------------------------------------------------------------------------
__global__ __launch_bounds__(256)
void rbm_local_op_kernel(const float* __restrict__ x, const float* __restrict__ a,
                         const float* __restrict__ Oxy, const float2* __restrict__ cs,
                         const float* __restrict__ T, float* __restrict__ out) {
    const int b  = blockIdx.y;
    const int kc = blockIdx.x;                 // 0..7
    __shared__ float Tl[Hcfg];
    __shared__ float bsum;
    for (int i = threadIdx.x; i < Hcfg; i += 256)
        Tl[i] = T[(size_t)b * Hcfg + i];
    if (threadIdx.x == 0) bsum = 0.0f;
    __syncthreads();

    const int wave = threadIdx.x >> 5;
    const int lane = threadIdx.x & 31;
    float acc = 0.0f;
    const int kend = kc * 64 + 64;
    for (int k = kc * 64 + wave; k < kend; k += 8) {
        const float xk = x[(size_t)b * Ncfg + k];
        const float2* __restrict__ row = cs + (size_t)k * Hcfg;
        float p0 = 1.0f, p1 = 1.0f;
#pragma unroll 4
        for (int h = lane; h < Hcfg; h += 64) {
            float2 v0 = row[h];
            float2 v1 = row[h + 32];
            p0 *= fmaf(-xk * v0.y, Tl[h],      v0.x);
            p1 *= fmaf(-xk * v1.y, Tl[h + 32], v1.x);
        }
        float p = p0 * p1;
#pragma unroll
        for (int off = 16; off > 0; off >>= 1)
            p *= __shfl_xor(p, off, 32);
        if (lane == 0)
            acc += Oxy[k] * p * expf(-2.0f * xk * a[k]);
    }
    if (lane == 0) atomicAdd(&bsum, acc);
    __syncthreads();
    if (threadIdx.x == 0) atomicAdd(&out[b], bsum);
}

// ---------------------------------------------------------------------------
// Launcher. Inputs: x[64,512], W[512,2048], b[2048], a[512], Oxy[512]; out[64] f32.
// Workspace: cs tables (N*H float2 = 8 MB) + tanh(theta) (B*H f32 = 0.5 MB).
// ---------------------------------------------------------------------------
extern "C" void kernel_launch(void* const* d_in, const int* in_sizes, int n_in,
                              void* d_out, int out_size, void* d_ws, size_t ws_size,
                              hipStream_t stream) {
    const float* x    = (const float*)d_in[0];
    const float* W    = (const float*)d_in[1];
    const float* bias = (const float*)d_in[2];
    const float* a    = (const float*)d_in[3];
    const float* Oxy  = (const float*)d_in[4];
    float* out        = (float*)d_out;

    float2* cs = (float2*)d_ws;                                             // 8 MB
    float*  T  = (float*)((char*)d_ws + (size_t)Ncfg * Hcfg * sizeof(float2)); // 0.5 MB

    hipMemsetAsync(d_out, 0, Bcfg * sizeof(float), stream);

    rbm_coshsinh_kernel<<<(Ncfg * Hcfg + 255) / 256, 256, 0, stream>>>(W, cs);
    rbm_theta_tanh_kernel<<<128, 128, 0, stream>>>(x, W, bias, T);
    rbm_local_op_kernel<<<dim3(8, Bcfg), 256, 0, stream>>>(x, a, Oxy, cs, T, out);
}